// RoICrop_52063593562789
// MI455X (gfx1250) — compile-verified
//
#include <hip/hip_runtime.h>

// Native clang vector types (HIP_vector_type float4 is rejected by
// __builtin_nontemporal_store; ext_vector_type works and lowers to b128).
typedef float v4f __attribute__((ext_vector_type(4)));
typedef int   v4i __attribute__((ext_vector_type(4)));

// Problem constants (from reference setup_inputs)
#define N_   128
#define C_   512
#define H_   38
#define W_   38
#define HO_  14
#define WO_  14
#define S_   (HO_*WO_)        // 196 samples per image (divisible by 4)
#define PLANE (H_*W_)         // 1444 floats per channel plane (5776 B, 16B-aligned)
#define CPB  8                // channel planes staged per block
#define CHUNK_B128 ((CPB*PLANE*4)/16)   // 2888 16-byte async transfers
#define NQUADS ((CPB*S_)/4)             // 392 output quads per block

__global__ __launch_bounds__(256) void roicrop_async_kernel(
    const float* __restrict__ feat,   // [N, C, H, W]
    const float* __restrict__ grid,   // [N, Ho, Wo, 2] (y, x)
    float* __restrict__ out)          // [N, C, Ho, Wo]
{
    __shared__ __align__(16) float planes[CPB * PLANE];  // 46208 B via async DMA
    __shared__ __align__(16) float wts[4][S_];           // corner weights (validity-zeroed)
    __shared__ __align__(16) int   adr[4][S_];           // clipped corner offsets (elements)

    const int tid = threadIdx.x;
    const int blk = blockIdx.x;
    const int nchunks = C_ / CPB;           // 64
    const int n  = blk / nchunks;
    const int c0 = (blk - n * nchunks) * CPB;

    // ---- Stage 0: per-sample bilinear params into LDS (threads 0..195) ----
    if (tid < S_) {
        const float yg = grid[((size_t)n * S_ + tid) * 2 + 0];
        const float xg = grid[((size_t)n * S_ + tid) * 2 + 1];
        const float xf = (xg + 1.0f) * 0.5f * (float)(W_ - 1);
        const float yf = (yg + 1.0f) * 0.5f * (float)(H_ - 1);
        const float x0f = floorf(xf), y0f = floorf(yf);
        const float wx = xf - x0f,    wy = yf - y0f;
        const int x0 = (int)x0f, y0 = (int)y0f;
        const int x1 = x0 + 1,   y1 = y0 + 1;
        const bool vx0 = (x0 >= 0) && (x0 < W_);
        const bool vx1 = (x1 >= 0) && (x1 < W_);
        const bool vy0 = (y0 >= 0) && (y0 < H_);
        const bool vy1 = (y1 >= 0) && (y1 < H_);
        const int xc0 = min(max(x0, 0), W_ - 1);
        const int xc1 = min(max(x1, 0), W_ - 1);
        const int yc0 = min(max(y0, 0), H_ - 1);
        const int yc1 = min(max(y1, 0), H_ - 1);
        wts[0][tid] = (1.0f - wy) * (1.0f - wx) * ((vy0 && vx0) ? 1.0f : 0.0f);
        wts[1][tid] = (1.0f - wy) * wx          * ((vy0 && vx1) ? 1.0f : 0.0f);
        wts[2][tid] = wy * (1.0f - wx)          * ((vy1 && vx0) ? 1.0f : 0.0f);
        wts[3][tid] = wy * wx                   * ((vy1 && vx1) ? 1.0f : 0.0f);
        adr[0][tid] = yc0 * W_ + xc0;
        adr[1][tid] = yc0 * W_ + xc1;
        adr[2][tid] = yc1 * W_ + xc0;
        adr[3][tid] = yc1 * W_ + xc1;
    }

    // ---- Stage 1: async DMA 8 contiguous channel planes (46208 B) -> LDS ----
    // Channels are contiguous within an image, so the 8 planes form one linear
    // 16B-aligned region. Each plane byte is read exactly once device-wide by
    // exactly one block -> non-temporal to keep the 378 MB stream out of L2.
    const char* gbase = (const char*)(feat + (size_t)(n * C_ + c0) * PLANE);
    // Low 32 bits of the generic address of an LDS object == LDS byte offset.
    const unsigned lbase = (unsigned)(uintptr_t)(void*)planes;
    for (int i = tid; i < CHUNK_B128; i += 256) {
        unsigned    loff = lbase + (unsigned)i * 16u;
        const char* g    = gbase + (size_t)i * 16u;
        asm volatile("global_load_async_to_lds_b128 %0, %1, off th:TH_LOAD_NT"
                     :: "v"(loff), "v"(g)
                     : "memory");
    }
    // Each wave waits for its own async transfers; the barrier then makes all
    // waves' LDS writes (async planes + stage-0 params) visible block-wide.
    asm volatile("s_wait_asynccnt 0x0" ::: "memory");
    __syncthreads();

    // ---- Stage 2: 392 quads (8 ch x 196 samples) from LDS, b128 stores ----
    // o = cl*196 + s is the linear offset into the contiguous output chunk for
    // (n, c0..c0+7); 196 % 4 == 0 so quads never cross a channel boundary.
    float* outbase = out + (size_t)(n * C_ + c0) * S_;
    for (int q = tid; q < NQUADS; q += 256) {
        const int o  = q * 4;
        const int cl = o / S_;
        const int s  = o - cl * S_;
        const float* pl = planes + cl * PLANE;

        const v4f w0 = *(const v4f*)&wts[0][s];
        const v4f w1 = *(const v4f*)&wts[1][s];
        const v4f w2 = *(const v4f*)&wts[2][s];
        const v4f w3 = *(const v4f*)&wts[3][s];
        const v4i i0 = *(const v4i*)&adr[0][s];
        const v4i i1 = *(const v4i*)&adr[1][s];
        const v4i i2 = *(const v4i*)&adr[2][s];
        const v4i i3 = *(const v4i*)&adr[3][s];

        v4f r;
        r.x = fmaf(w3.x, pl[i3.x], fmaf(w2.x, pl[i2.x], fmaf(w1.x, pl[i1.x], w0.x * pl[i0.x])));
        r.y = fmaf(w3.y, pl[i3.y], fmaf(w2.y, pl[i2.y], fmaf(w1.y, pl[i1.y], w0.y * pl[i0.y])));
        r.z = fmaf(w3.z, pl[i3.z], fmaf(w2.z, pl[i2.z], fmaf(w1.z, pl[i1.z], w0.z * pl[i0.z])));
        r.w = fmaf(w3.w, pl[i3.w], fmaf(w2.w, pl[i2.w], fmaf(w1.w, pl[i1.w], w0.w * pl[i0.w])));

        // Write-once output: non-temporal 16B store.
        __builtin_nontemporal_store(r, (v4f*)(outbase + o));
    }
}

extern "C" void kernel_launch(void* const* d_in, const int* in_sizes, int n_in,
                              void* d_out, int out_size, void* d_ws, size_t ws_size,
                              hipStream_t stream) {
    (void)in_sizes; (void)n_in; (void)out_size; (void)d_ws; (void)ws_size;
    const float* feat = (const float*)d_in[0];   // [128,512,38,38] fp32
    const float* grid = (const float*)d_in[1];   // [128,14,14,2]   fp32
    float*       out  = (float*)d_out;           // [128,512,14,14] fp32

    const int blocks = N_ * (C_ / CPB);          // 8192 blocks x 256 threads
    roicrop_async_kernel<<<blocks, 256, 0, stream>>>(feat, grid, out);
}